// GraphNet_82806969467523
// MI455X (gfx1250) — compile-verified
//
#include <hip/hip_runtime.h>

// GCN forward, MI455X (gfx1250, wave32).
//   N=50000 nodes, E=640000 edges, D=H=128, G=64 graphs, O=16.
// Dense GEMMs use V_WMMA_F32_16X16X4_F32 (fp32, matches reference dtype).
// Sparse aggregation: one wave per edge, native fp32 global atomics
// (25.6MB scatter target lives in the 192MB L2). W staged in LDS.

#define NN   50000
#define EE   640000
#define DIM  128
#define GG   64
#define OO   16

typedef __attribute__((ext_vector_type(2))) float v2f;
typedef __attribute__((ext_vector_type(8))) float v8f;

// Native fp32 global atomic add (GLOBAL_ATOMIC_ADD_F32, no CAS loop).
// Targets are coarse-grained device workspace memory, so this is safe.
__device__ __forceinline__ void atomic_add_f32(float* p, float v) {
  unsafeAtomicAdd(p, v);
}

// ---------------- degree / norm ----------------
__global__ __launch_bounds__(256) void k_init_deg(float* deg, int n) {
  int i = blockIdx.x * 256 + threadIdx.x;
  if (i < n) deg[i] = 1.0f;                    // self-loop contributes weight 1
}

__global__ __launch_bounds__(256) void k_deg_acc(const int* __restrict__ col,
                                                 const float* __restrict__ ew,
                                                 float* deg, int e) {
  int i = blockIdx.x * 256 + threadIdx.x;
  if (i < e) atomic_add_f32(&deg[col[i]], ew[i]);
}

__global__ __launch_bounds__(256) void k_dinv(float* deg, int n) {
  int i = blockIdx.x * 256 + threadIdx.x;
  if (i < n) {
    float d = deg[i];
    deg[i] = (d > 0.0f) ? rsqrtf(d) : 0.0f;    // deg >= 1 always (self loop)
  }
}

// ---------------- zero fill ----------------
__global__ __launch_bounds__(256) void k_zero(float4* p, size_t n4) {
  size_t i = (size_t)blockIdx.x * 256 + threadIdx.x;
  if (i < n4) p[i] = make_float4(0.f, 0.f, 0.f, 0.f);
}

// ---------------- fp32 WMMA GEMM: out[N,128] = A[N,128] @ W[128,128] ------
// Block = 256 threads = 8 waves; each wave owns a 16-row strip, computing the
// full 16x128 output via 8 v8f accumulators. K loop: 32 steps of K=4.
// Fragment layouts per CDNA5 ISA 7.12.2 (32-bit A 16x4 / B 4x16 / C 16x16).
__global__ __launch_bounds__(256) void k_gemm_wmma(const float* __restrict__ A,
                                                   const float* __restrict__ W,
                                                   float* __restrict__ out,
                                                   int nrows) {
  __shared__ float w_lds[DIM * DIM];           // 64 KB of 320 KB/WGP
  // cooperative load of W into LDS (float4: 4096 vec4 / 256 threads)
  const float4* wsrc = (const float4*)W;
  float4* wdst = (float4*)w_lds;
  #pragma unroll
  for (int i = 0; i < (DIM * DIM / 4) / 256; ++i)
    wdst[threadIdx.x + i * 256] = wsrc[threadIdx.x + i * 256];
  __syncthreads();

  const int wave  = threadIdx.x >> 5;
  const int lane  = threadIdx.x & 31;
  const int l16   = lane & 15;                 // M (A/C) or N (B/C) sub-index
  const int khalf = lane >> 4;                 // selects K pair (0 or 1)

  const int m0 = (blockIdx.x * 8 + wave) * 16;
  if (m0 >= nrows) return;                     // whole-wave guard: EXEC all-1s

  v8f acc[8];
  #pragma unroll
  for (int t = 0; t < 8; ++t) acc[t] = (v8f){0.f,0.f,0.f,0.f,0.f,0.f,0.f,0.f};

  const float* arow = A + (size_t)(m0 + l16) * DIM;
  for (int k = 0; k < DIM; k += 4) {
    const int ka = k + 2 * khalf;              // even -> 8B-aligned float2
    v2f a = *(const v2f*)(arow + ka);
    #pragma unroll
    for (int t = 0; t < 8; ++t) {
      const int n0 = t * 16;
      v2f b;
      b.x = w_lds[ka * DIM + n0 + l16];        // B[K=ka  ][N=n0+l16]
      b.y = w_lds[(ka + 1) * DIM + n0 + l16];  // B[K=ka+1][N=n0+l16]
      acc[t] = __builtin_amdgcn_wmma_f32_16x16x4_f32(
          false, a, false, b, (short)0, acc[t], false, false);
    }
  }

  // C layout: VGPR r holds M = r + 8*khalf, N = l16
  #pragma unroll
  for (int t = 0; t < 8; ++t) {
    #pragma unroll
    for (int r = 0; r < 8; ++r) {
      const int m = m0 + r + 8 * khalf;
      out[(size_t)m * DIM + t * 16 + l16] = acc[t][r];
    }
  }
}

// ---------------- edge scatter: out[col] += norm * xw[row] ----------------
// One wave per edge; 4 features per lane (float4 gather, 4 fp32 atomics).
__global__ __launch_bounds__(256) void k_scatter(const float* __restrict__ xw,
                                                 const int* __restrict__ row,
                                                 const int* __restrict__ col,
                                                 const float* __restrict__ ew,
                                                 const float* __restrict__ dinv,
                                                 float* out, int e) {
  const int wave = threadIdx.x >> 5;
  const int lane = threadIdx.x & 31;
  const int idx = blockIdx.x * 8 + wave;
  if (idx >= e) return;
  const int r = row[idx];
  const int c = col[idx];
  const float nrm = dinv[r] * ew[idx] * dinv[c];
  const float4 v = *(const float4*)(xw + (size_t)r * DIM + lane * 4);
  float* dst = out + (size_t)c * DIM + lane * 4;
  atomic_add_f32(dst + 0, v.x * nrm);
  atomic_add_f32(dst + 1, v.y * nrm);
  atomic_add_f32(dst + 2, v.z * nrm);
  atomic_add_f32(dst + 3, v.w * nrm);
}

// ---- finalize: h = relu(agg + xw*dinv^2 (self-loop) + bias), in place ----
__global__ __launch_bounds__(256) void k_finalize(float* agg,
                                                  const float* __restrict__ xw,
                                                  const float* __restrict__ dinv,
                                                  const float* __restrict__ bias,
                                                  int total) {
  int i = blockIdx.x * 256 + threadIdx.x;
  if (i < total) {
    const int nd = i >> 7;        // /128
    const int f  = i & 127;
    const float di = dinv[nd];
    float v = agg[i] + xw[i] * di * di + bias[f];
    agg[i] = v > 0.f ? v : 0.f;
  }
}

// ---------------- mean-pool accumulate ----------------
__global__ __launch_bounds__(256) void k_pool(const float* __restrict__ h,
                                              const int* __restrict__ batch,
                                              float* pool, float* cnt, int n) {
  const int wave = threadIdx.x >> 5;
  const int lane = threadIdx.x & 31;
  const int nd = blockIdx.x * 8 + wave;
  if (nd >= n) return;
  const int g = batch[nd];
  const float4 v = *(const float4*)(h + (size_t)nd * DIM + lane * 4);
  float* dst = pool + (size_t)g * DIM + lane * 4;
  atomic_add_f32(dst + 0, v.x);
  atomic_add_f32(dst + 1, v.y);
  atomic_add_f32(dst + 2, v.z);
  atomic_add_f32(dst + 3, v.w);
  if (lane == 0) atomic_add_f32(&cnt[g], 1.0f);
}

// ---------------- final FC: out[G,16] = (pool/cnt) @ Wfc + bfc ------------
__global__ __launch_bounds__(256) void k_fc(const float* __restrict__ pool,
                                            const float* __restrict__ cnt,
                                            const float* __restrict__ Wfc,
                                            const float* __restrict__ bfc,
                                            float* out) {
  int i = blockIdx.x * 256 + threadIdx.x;
  if (i >= GG * OO) return;
  const int g = i / OO;
  const int o = i % OO;
  float acc = 0.f;
  #pragma unroll 8
  for (int f = 0; f < DIM; ++f) acc += pool[g * DIM + f] * Wfc[f * OO + o];
  float c = cnt[g];
  c = c > 1.f ? c : 1.f;
  out[i] = acc / c + bfc[o];
}

extern "C" void kernel_launch(void* const* d_in, const int* in_sizes, int n_in,
                              void* d_out, int out_size, void* d_ws, size_t ws_size,
                              hipStream_t stream) {
  const float* x     = (const float*)d_in[0];
  const int*   eidx  = (const int*)d_in[1];    // [2, E]
  const float* ew    = (const float*)d_in[2];
  const int*   batch = (const int*)d_in[3];
  const float* W1    = (const float*)d_in[4];
  const float* b1    = (const float*)d_in[5];
  const float* W2    = (const float*)d_in[6];
  const float* b2    = (const float*)d_in[7];
  const float* Wfc   = (const float*)d_in[8];
  const float* bfc   = (const float*)d_in[9];
  float* out = (float*)d_out;

  const int* row = eidx;            // edge_index[0]
  const int* col = eidx + EE;       // edge_index[1]

  // workspace layout (floats): deg/dinv | buf0 (N*128) | buf1 (N*128) | pool | cnt
  float* ws   = (float*)d_ws;
  float* deg  = ws;                                   // 50000 (padded 50048)
  float* buf0 = ws + 50048;
  float* buf1 = buf0 + (size_t)NN * DIM;
  float* pool = buf1 + (size_t)NN * DIM;
  float* cnt  = pool + GG * DIM;                      // contiguous after pool

  const int nThreads = 256;
  const int gN    = (NN + 255) / 256;                 // node-wise
  const int gE    = (EE + 255) / 256;                 // edge-wise (thread/edge)
  const int gGemm = (NN / 16 + 7) / 8;                // 3125 strips, 8 waves/blk
  const int gEw   = EE / 8;                           // wave per edge
  const int gNw   = NN / 8;                           // wave per node
  const int gZero = (int)(((size_t)NN * DIM / 4 + 255) / 256);
  const int gFin  = NN * DIM / 256;

  // degree -> dinv (stored in deg)
  k_init_deg<<<gN, nThreads, 0, stream>>>(deg, NN);
  k_deg_acc <<<gE, nThreads, 0, stream>>>(col, ew, deg, EE);
  k_dinv    <<<gN, nThreads, 0, stream>>>(deg, NN);

  // layer 1: buf0 = x@W1 ; buf1 = relu(scatter + self + b1)
  k_gemm_wmma<<<gGemm, nThreads, 0, stream>>>(x, W1, buf0, NN);
  k_zero     <<<gZero, nThreads, 0, stream>>>((float4*)buf1, (size_t)NN * DIM / 4);
  k_scatter  <<<gEw, nThreads, 0, stream>>>(buf0, row, col, ew, deg, buf1, EE);
  k_finalize <<<gFin, nThreads, 0, stream>>>(buf1, buf0, deg, b1, NN * DIM);

  // layer 2: buf0 = h1@W2 ; buf1 = relu(scatter + self + b2)
  k_gemm_wmma<<<gGemm, nThreads, 0, stream>>>(buf1, W2, buf0, NN);
  k_zero     <<<gZero, nThreads, 0, stream>>>((float4*)buf1, (size_t)NN * DIM / 4);
  k_scatter  <<<gEw, nThreads, 0, stream>>>(buf0, row, col, ew, deg, buf1, EE);
  k_finalize <<<gFin, nThreads, 0, stream>>>(buf1, buf0, deg, b2, NN * DIM);

  // pooling + FC
  const size_t poolN4 = (GG * DIM + GG) / 4;          // pool + cnt, contiguous
  k_zero<<<(int)((poolN4 + 255) / 256), nThreads, 0, stream>>>((float4*)pool, poolN4);
  k_pool<<<gNw, nThreads, 0, stream>>>(buf1, batch, pool, cnt, NN);
  k_fc  <<<(GG * OO + 255) / 256, nThreads, 0, stream>>>(pool, cnt, Wfc, bfc, out);
}